// TransactionLayer_35656818491725
// MI455X (gfx1250) — compile-verified
//
#include <hip/hip_runtime.h>

// Global spatial max pool: in [B=64, H=112, W=112, C=256] f32 (NHWC) -> out [B, C] f32
// Memory-bound: 822 MB streamed once -> ~35 us floor @ 23.3 TB/s.
// Strategy: 128-bit non-temporal coalesced loads, register max accumulation,
// LDS partial reduce per block, global_atomic_max_num_f32 combine across blocks.

typedef float v4f __attribute__((ext_vector_type(4)));

#define BATCH   64
#define HW      12544        // 112*112
#define CH      256
#define GROUPS  3136         // HW / 4 spatial positions per 4KB group
#define SCHUNKS 16           // spatial chunks per batch
#define GPB     (GROUPS / SCHUNKS)   // 196 groups per block

__global__ __launch_bounds__(256) void init_out_kernel(float* __restrict__ out, int n) {
    int i = blockIdx.x * 256 + threadIdx.x;
    if (i < n) out[i] = -__builtin_inff();
}

static __device__ __forceinline__ v4f vmax4(v4f a, v4f b) {
    v4f r;
    r.x = __builtin_fmaxf(a.x, b.x);
    r.y = __builtin_fmaxf(a.y, b.y);
    r.z = __builtin_fmaxf(a.z, b.z);
    r.w = __builtin_fmaxf(a.w, b.w);
    return r;
}

__global__ __launch_bounds__(256) void maxpool_kernel(const float* __restrict__ in,
                                                      float* __restrict__ out) {
    const int tid   = threadIdx.x;          // 0..255 (8 waves, wave32)
    const int chunk = blockIdx.x;           // 0..SCHUNKS-1
    const int b     = blockIdx.y;           // 0..BATCH-1

    // Block's first spatial position = chunk*GPB*4; lane covers 4 fixed channels:
    // channel slot = (tid*4) % 256, spatial partition = tid/64 within each group.
    const float* p0 = in + ((size_t)b * HW + (size_t)chunk * (GPB * 4)) * CH + (size_t)tid * 4;

    v4f a0, a1, a2, a3;
    a0 = a1 = a2 = a3 = (v4f){-__builtin_inff(), -__builtin_inff(),
                              -__builtin_inff(), -__builtin_inff()};

    // GPB = 196 = 4 * 49 groups; stride per group = 4 positions * 256 ch = 1024 floats (4 KB).
    #pragma unroll 1
    for (int g = 0; g < GPB; g += 4) {
        const float* p = p0 + (size_t)g * (4 * CH);
        // Prefetch ~8 groups ahead into the cache hierarchy (global_prefetch_b8).
        __builtin_prefetch(p + (size_t)8 * (4 * CH), 0, 0);
        // Four independent 128-bit NT loads -> 4 outstanding vmem ops per lane.
        v4f v0 = __builtin_nontemporal_load((const v4f*)(p + 0 * (4 * CH)));
        v4f v1 = __builtin_nontemporal_load((const v4f*)(p + 1 * (4 * CH)));
        v4f v2 = __builtin_nontemporal_load((const v4f*)(p + 2 * (4 * CH)));
        v4f v3 = __builtin_nontemporal_load((const v4f*)(p + 3 * (4 * CH)));
        a0 = vmax4(a0, v0);
        a1 = vmax4(a1, v1);
        a2 = vmax4(a2, v2);
        a3 = vmax4(a3, v3);
    }
    v4f acc = vmax4(vmax4(a0, a1), vmax4(a2, a3));

    // Threads {t, t+64, t+128, t+192} hold the same 4 channels for different
    // spatial partitions: reduce across them through LDS (4 KB).
    __shared__ v4f sm[256];
    sm[tid] = acc;
    __syncthreads();

    if (tid < 64) {
        v4f r = vmax4(vmax4(sm[tid], sm[tid + 64]),
                      vmax4(sm[tid + 128], sm[tid + 192]));
        float* o = out + (size_t)b * CH + (size_t)tid * 4;
        // CDNA5 float atomic max (global_atomic_max_num_f32, no-return form).
        (void)__hip_atomic_fetch_max(o + 0, r.x, __ATOMIC_RELAXED, __HIP_MEMORY_SCOPE_AGENT);
        (void)__hip_atomic_fetch_max(o + 1, r.y, __ATOMIC_RELAXED, __HIP_MEMORY_SCOPE_AGENT);
        (void)__hip_atomic_fetch_max(o + 2, r.z, __ATOMIC_RELAXED, __HIP_MEMORY_SCOPE_AGENT);
        (void)__hip_atomic_fetch_max(o + 3, r.w, __ATOMIC_RELAXED, __HIP_MEMORY_SCOPE_AGENT);
    }
}

extern "C" void kernel_launch(void* const* d_in, const int* in_sizes, int n_in,
                              void* d_out, int out_size, void* d_ws, size_t ws_size,
                              hipStream_t stream) {
    (void)in_sizes; (void)n_in; (void)d_ws; (void)ws_size;
    const float* in = (const float*)d_in[0];
    float* out = (float*)d_out;

    // Re-initialize output to -inf every call (graph replays don't re-poison).
    init_out_kernel<<<(out_size + 255) / 256, 256, 0, stream>>>(out, out_size);

    dim3 grid(SCHUNKS, BATCH);   // 1024 blocks x 8 waves
    maxpool_kernel<<<grid, 256, 0, stream>>>(in, out);
}